// GAT_85676007621064
// MI455X (gfx1250) — compile-verified
//
#include <hip/hip_runtime.h>
#include <math.h>

// ---------------------------------------------------------------------------
// GAT (2x GATConv + linear head) for MI455X / gfx1250, wave32.
// GEMMs use V_WMMA_F32_16X16X4_F32 (exact fp32, matches reference precision),
// 32-row strips per block so each B fragment feeds two WMMAs.
// Edge softmax/aggregation via global f32/u32 atomics (L2-resident accum).
// ---------------------------------------------------------------------------

typedef float v2f __attribute__((ext_vector_type(2)));
typedef float v8f __attribute__((ext_vector_type(8)));

#define NEG_SLOPE 0.2f

// Order-preserving map float -> uint (for atomicMax-based segment max).
__device__ __forceinline__ unsigned fkey(float f) {
  unsigned u = __float_as_uint(f);
  return (u & 0x80000000u) ? ~u : (u | 0x80000000u);
}
__device__ __forceinline__ float fval(unsigned k) {
  unsigned u = (k & 0x80000000u) ? (k ^ 0x80000000u) : ~k;
  return __uint_as_float(u);
}

// Store one 16x16 C/D tile (8 VGPRs; rows i + (lane>=16 ? 8 : 0), col=lane%16).
// Uniform fast path when the whole tile is in range (always true except the
// final strip when M % 32 != 0).
__device__ __forceinline__ void store_tile(float* __restrict__ Y, int M,
                                           int t0, int hi, int col, v8f acc) {
  const int rbase = t0 + (hi << 3);
  if (t0 + 16 <= M) {
#pragma unroll
    for (int i = 0; i < 8; ++i) Y[(size_t)(rbase + i) * 128 + col] = acc[i];
  } else {
#pragma unroll
    for (int i = 0; i < 8; ++i) {
      const int r = rbase + i;
      if (r < M) Y[(size_t)r * 128 + col] = acc[i];
    }
  }
}

// ---------------------------------------------------------------------------
// Y[M,128] = X[M,128] @ W[128,128] using f32 WMMA 16x16x4.
// grid.x = ceil(M/32); 8 waves/block, wave w owns output cols 16w..16w+15 for
// two stacked 16-row tiles (B fragment reused across both).
// A-frag: lane L%16 = row, VGPR0/1 = K{0,1} (lanes<16) or K{2,3} (lanes>=16);
// B-frag mirrored.
// ---------------------------------------------------------------------------
__global__ __launch_bounds__(256) void gemm128_wmma(
    const float* __restrict__ X, const float* __restrict__ W,
    float* __restrict__ Y, int M) {
  const int lane  = threadIdx.x & 31;
  const int wave  = threadIdx.x >> 5;
  const int m0    = blockIdx.x << 5;    // 32-row strip
  const int l16   = lane & 15;
  const int hi    = lane >> 4;          // 0: lanes 0-15, 1: lanes 16-31
  const int khalf = hi << 1;            // 0 or 2
  const int col   = (wave << 4) + l16;  // output/B column, < 128
  int arow0 = m0 + l16;
  int arow1 = m0 + 16 + l16;
  if (arow0 > M - 1) arow0 = M - 1;     // clamp (EXEC must stay all-1 for WMMA)
  if (arow1 > M - 1) arow1 = M - 1;

  const float* __restrict__ xr0 = X + (size_t)arow0 * 128;
  const float* __restrict__ xr1 = X + (size_t)arow1 * 128;
  v8f acc0 = {}, acc1 = {};
#pragma unroll 4
  for (int k = 0; k < 128; k += 4) {
    v2f b, a0, a1;
    b.x  = W[(size_t)(k + khalf) * 128 + col];
    b.y  = W[(size_t)(k + khalf + 1) * 128 + col];
    a0.x = xr0[k + khalf];
    a0.y = xr0[k + khalf + 1];
    a1.x = xr1[k + khalf];
    a1.y = xr1[k + khalf + 1];
    acc0 = __builtin_amdgcn_wmma_f32_16x16x4_f32(
        false, a0, false, b, (short)0, acc0, false, false);
    acc1 = __builtin_amdgcn_wmma_f32_16x16x4_f32(
        false, a1, false, b, (short)0, acc1, false, false);
  }
  store_tile(Y, M, m0, hi, col, acc0);
  store_tile(Y, M, m0 + 16, hi, col, acc1);
}

// ---------------------------------------------------------------------------
// Per (node,head): attention logit halves + init max/denom + zero accumulator.
// ---------------------------------------------------------------------------
__global__ __launch_bounds__(256) void node_alpha(
    const float* __restrict__ XW, const float* __restrict__ a_src,
    const float* __restrict__ a_dst, float* __restrict__ asrc,
    float* __restrict__ adst, unsigned* __restrict__ mkey,
    float* __restrict__ denom, float* __restrict__ outacc, int n) {
  const int t = blockIdx.x * blockDim.x + threadIdx.x;
  if (t >= n * 4) return;
  const int node = t >> 2, h = t & 3;
  const float* __restrict__ row = XW + (size_t)node * 128 + h * 32;
  const float* __restrict__ vs = a_src + h * 32;
  const float* __restrict__ vd = a_dst + h * 32;
  float ss = 0.f, sd = 0.f;
#pragma unroll 8
  for (int c = 0; c < 32; ++c) {
    const float v = row[c];
    ss = fmaf(v, vs[c], ss);
    sd = fmaf(v, vd[c], sd);
  }
  asrc[t] = ss;
  adst[t] = sd;
  mkey[t] = 0u;     // < fkey(any finite); self-loop guarantees overwrite
  denom[t] = 0.f;
  float* __restrict__ orow = outacc + (size_t)node * 128 + h * 32;
#pragma unroll 8
  for (int c = 0; c < 32; ++c) orow[c] = 0.f;
}

__device__ __forceinline__ void edge_sd(const int* __restrict__ ei, int E,
                                        int e, int& s, int& d) {
  if (e < E) { s = ei[e]; d = ei[E + e]; }
  else       { s = d = e - E; }            // self-loops appended
}

// Pass 1: per-dst segment max of leaky-relu logits (one thread per edge*head).
__global__ __launch_bounds__(256) void edge_max(
    const int* __restrict__ ei, int E, int n, const float* __restrict__ asrc,
    const float* __restrict__ adst, unsigned* __restrict__ mkey) {
  const int t = blockIdx.x * blockDim.x + threadIdx.x;
  if (t >= (E + n) * 4) return;
  const int e = t >> 2, h = t & 3;
  int s, d; edge_sd(ei, E, e, s, d);
  float el = asrc[s * 4 + h] + adst[d * 4 + h];
  el = (el > 0.f) ? el : NEG_SLOPE * el;
  atomicMax(&mkey[d * 4 + h], fkey(el));
}

// Pass 2: per-dst segment sum of exp(logit - max).
__global__ __launch_bounds__(256) void edge_sum(
    const int* __restrict__ ei, int E, int n, const float* __restrict__ asrc,
    const float* __restrict__ adst, const unsigned* __restrict__ mkey,
    float* __restrict__ denom) {
  const int t = blockIdx.x * blockDim.x + threadIdx.x;
  if (t >= (E + n) * 4) return;
  const int e = t >> 2, h = t & 3;
  int s, d; edge_sd(ei, E, e, s, d);
  float el = asrc[s * 4 + h] + adst[d * 4 + h];
  el = (el > 0.f) ? el : NEG_SLOPE * el;
  atomicAdd(&denom[d * 4 + h], expf(el - fval(mkey[d * 4 + h])));
}

// Pass 3: weighted gather/scatter. One wave per edge; lane covers channel
// h*32 + lane%32 for each head h (coalesced 128B per head per wave).
__global__ __launch_bounds__(256) void edge_agg(
    const int* __restrict__ ei, int E, int n, const float* __restrict__ asrc,
    const float* __restrict__ adst, const unsigned* __restrict__ mkey,
    const float* __restrict__ denom, const float* __restrict__ XW,
    float* __restrict__ outacc) {
  const int wid = (blockIdx.x * blockDim.x + threadIdx.x) >> 5;
  const int lane = threadIdx.x & 31;
  if (wid >= E + n) return;
  int s, d; edge_sd(ei, E, wid, s, d);
  const float* __restrict__ xr = XW + (size_t)s * 128;
  float* __restrict__ orow = outacc + (size_t)d * 128;
#pragma unroll
  for (int h = 0; h < 4; ++h) {
    float el = asrc[s * 4 + h] + adst[d * 4 + h];      // scalar broadcast loads
    el = (el > 0.f) ? el : NEG_SLOPE * el;
    const float alpha = expf(el - fval(mkey[d * 4 + h])) / denom[d * 4 + h];
    const int c = h * 32 + lane;
    atomicAdd(&orow[c], alpha * xr[c]);
  }
}

// out[t] = elu(acc[t] + b[t%128])  (in place)
__global__ __launch_bounds__(256) void bias_elu(
    float* __restrict__ acc, const float* __restrict__ b, int n) {
  const int t = blockIdx.x * blockDim.x + threadIdx.x;
  if (t >= n * 128) return;
  const float v = acc[t] + b[t & 127];
  acc[t] = (v > 0.f) ? v : expm1f(v);
}

// Regression head: one wave per node, 128-wide dot + wave32 reduction.
__global__ __launch_bounds__(256) void head_dot(
    const float* __restrict__ Hf, const float* __restrict__ Wr,
    const float* __restrict__ br, float* __restrict__ out, int n) {
  const int wid = (blockIdx.x * blockDim.x + threadIdx.x) >> 5;
  const int lane = threadIdx.x & 31;
  if (wid >= n) return;
  const float* __restrict__ row = Hf + (size_t)wid * 128;
  float s = 0.f;
#pragma unroll
  for (int c = 0; c < 4; ++c) s = fmaf(row[lane + 32 * c], Wr[lane + 32 * c], s);
#pragma unroll
  for (int off = 16; off > 0; off >>= 1) s += __shfl_xor(s, off, 32);
  if (lane == 0) out[wid] = s + br[0];
}

// ---------------------------------------------------------------------------
extern "C" void kernel_launch(void* const* d_in, const int* in_sizes, int n_in,
                              void* d_out, int out_size, void* d_ws,
                              size_t ws_size, hipStream_t stream) {
  const float* x   = (const float*)d_in[0];
  const int*   ei  = (const int*)d_in[1];
  const float* W0  = (const float*)d_in[2];
  const float* as0 = (const float*)d_in[3];
  const float* ad0 = (const float*)d_in[4];
  const float* b0  = (const float*)d_in[5];
  const float* W1  = (const float*)d_in[6];
  const float* as1 = (const float*)d_in[7];
  const float* ad1 = (const float*)d_in[8];
  const float* b1  = (const float*)d_in[9];
  const float* Wr  = (const float*)d_in[10];
  const float* br  = (const float*)d_in[11];
  float* out = (float*)d_out;

  const int n = in_sizes[0] / 128;   // 50000
  const int E = in_sizes[1] / 2;     // 800000
  const int EN = E + n;

  // Workspace layout (~55 MB): two feature buffers + per-(node,head) arrays.
  float* bufA   = (float*)d_ws;                    // XW scratch   [n,128]
  float* bufB   = bufA + (size_t)n * 128;          // feat/accum   [n,128]
  float* asrc   = bufB + (size_t)n * 128;          // [n,4]
  float* adst   = asrc + (size_t)n * 4;            // [n,4]
  float* denom  = adst + (size_t)n * 4;            // [n,4]
  unsigned* mkey = (unsigned*)(denom + (size_t)n * 4); // [n,4]

  const int gM   = (n + 31) / 32;    // 32-row strips
  const int gNH  = (n * 4 + 255) / 256;
  const int gEH  = (EN * 4 + 255) / 256;
  const int gEW  = (EN + 7) / 8;     // 8 waves/block
  const int gNC  = (n * 128 + 255) / 256;
  const int gHD  = (n + 7) / 8;

  // ---- Layer 0 ----
  gemm128_wmma<<<gM, 256, 0, stream>>>(x, W0, bufA, n);
  node_alpha<<<gNH, 256, 0, stream>>>(bufA, as0, ad0, asrc, adst, mkey, denom, bufB, n);
  edge_max<<<gEH, 256, 0, stream>>>(ei, E, n, asrc, adst, mkey);
  edge_sum<<<gEH, 256, 0, stream>>>(ei, E, n, asrc, adst, mkey, denom);
  edge_agg<<<gEW, 256, 0, stream>>>(ei, E, n, asrc, adst, mkey, denom, bufA, bufB);
  bias_elu<<<gNC, 256, 0, stream>>>(bufB, b0, n);

  // ---- Layer 1 ----
  gemm128_wmma<<<gM, 256, 0, stream>>>(bufB, W1, bufA, n);
  node_alpha<<<gNH, 256, 0, stream>>>(bufA, as1, ad1, asrc, adst, mkey, denom, bufB, n);
  edge_max<<<gEH, 256, 0, stream>>>(ei, E, n, asrc, adst, mkey);
  edge_sum<<<gEH, 256, 0, stream>>>(ei, E, n, asrc, adst, mkey, denom);
  edge_agg<<<gEW, 256, 0, stream>>>(ei, E, n, asrc, adst, mkey, denom, bufA, bufB);
  bias_elu<<<gNC, 256, 0, stream>>>(bufB, b1, n);

  // ---- Head ----
  head_dot<<<gHD, 256, 0, stream>>>(bufB, Wr, br, out, n);
}